// PostCountPredictor_36850819400390
// MI455X (gfx1250) — compile-verified
//
#include <hip/hip_runtime.h>

#define N_NODES 8192
#define N_EDGES 8192
#define DEG 32
#define NNZ_TOT (N_EDGES * DEG)
#define D 32

typedef float v2f __attribute__((ext_vector_type(2)));
typedef float v8f __attribute__((ext_vector_type(8)));

// ---------------------------------------------------------------------------
// Collapse the activation-free MLP: Weff = Wm1@Wm2@Wm3 (64x1),
// beff = bm1@Wm2@Wm3 + bm2@Wm3 + bm3.
// ---------------------------------------------------------------------------
__global__ void k_build_weff(const float* __restrict__ Wm1, const float* __restrict__ bm1,
                             const float* __restrict__ Wm2, const float* __restrict__ bm2,
                             const float* __restrict__ Wm3, const float* __restrict__ bm3,
                             float* __restrict__ weff, float* __restrict__ beff) {
  __shared__ float w23[D];
  int t = threadIdx.x;
  if (t < D) {
    float s = 0.f;
    for (int k = 0; k < D; ++k) s += Wm2[t * D + k] * Wm3[k];
    w23[t] = s;
  }
  __syncthreads();
  if (t < 2 * D) {
    float s = 0.f;
    for (int j = 0; j < D; ++j) s += Wm1[t * D + j] * w23[j];
    weff[t] = s;
  }
  if (t == 0) {
    float s = bm3[0];
    for (int j = 0; j < D; ++j) s += bm1[j] * w23[j];
    for (int k = 0; k < D; ++k) s += bm2[k] * Wm3[k];
    beff[0] = s;
  }
}

// ---------------------------------------------------------------------------
// Edge aggregation: x_edge[e,:] = sum over UNIQUE nodes of this hyperedge.
// One wave32 per edge; lane doubles as (a) holder of node index #lane and
// (b) accumulator of feature column #lane. Dedup via shfl + ballot.
// ---------------------------------------------------------------------------
__global__ void __launch_bounds__(256) k_edge_agg(const int* __restrict__ node_idx,
                                                  const float* __restrict__ xnode,
                                                  float* __restrict__ xedge) {
  int lane = threadIdx.x & 31;
  int e = (blockIdx.x * blockDim.x + threadIdx.x) >> 5;
  int myIdx = node_idx[e * DEG + lane];
  bool dup = false;
#pragma unroll
  for (int j = 0; j < 32; ++j) {
    int vj = __shfl(myIdx, j, 32);
    if (j < lane && vj == myIdx) dup = true;
  }
  unsigned keep = ~(unsigned)__ballot(dup);
  float acc = 0.f;
#pragma unroll
  for (int k = 0; k < 32; ++k) {
    int idxk = __shfl(myIdx, k, 32);
    float v = xnode[idxk * D + lane];
    acc += ((keep >> k) & 1u) ? v : 0.f;
  }
  xedge[e * D + lane] = acc;
}

// ---------------------------------------------------------------------------
// Node scatter: x_node[n,:] += t_edge[e,:] for each unique (n,e) pair.
// One wave32 per edge, atomic f32 adds (global_atomic_add_f32).
// ---------------------------------------------------------------------------
__global__ void __launch_bounds__(256) k_node_scatter(const int* __restrict__ node_idx,
                                                      const float* __restrict__ tedge,
                                                      float* __restrict__ xnode) {
  int lane = threadIdx.x & 31;
  int e = (blockIdx.x * blockDim.x + threadIdx.x) >> 5;
  int myIdx = node_idx[e * DEG + lane];
  bool dup = false;
#pragma unroll
  for (int j = 0; j < 32; ++j) {
    int vj = __shfl(myIdx, j, 32);
    if (j < lane && vj == myIdx) dup = true;
  }
  unsigned keep = ~(unsigned)__ballot(dup);
  float val = tedge[e * D + lane];
#pragma unroll
  for (int k = 0; k < 32; ++k) {
    int idxk = __shfl(myIdx, k, 32);
    if ((keep >> k) & 1u) atomicAdd(&xnode[idxk * D + lane], val);
  }
}

// ---------------------------------------------------------------------------
// Y[E,32] = X[E,32] @ W[32,32] via V_WMMA_F32_16X16X4_F32 (exact f32 path).
// One wave per 16-row slab; 8 K-steps x 2 N-halves = 16 WMMA per wave.
// Layouts per ISA 7.12.2:
//   A 16x4 : lane m=lane&15; VGPR0 -> K = base+{0|2}, VGPR1 -> K = base+{1|3}
//   B 4x16 : lane n=lane&15; mirrored K striping
//   C/D    : VGPR v -> row M = v + (lane<16 ? 0 : 8), col N = lane&15
// ---------------------------------------------------------------------------
__global__ void __launch_bounds__(256) k_edge_linear_wmma(const float* __restrict__ X,
                                                          const float* __restrict__ W,
                                                          float* __restrict__ Y) {
  int lane = threadIdx.x & 31;
  int wave = (blockIdx.x * blockDim.x + threadIdx.x) >> 5;
  int rowBase = wave * 16;
  int mn = lane & 15;              // M for A / N for B,D
  int hi = lane >> 4;              // 0: lanes 0-15, 1: lanes 16-31
  int koff = hi ? 2 : 0;

  v8f acc0 = {};
  v8f acc1 = {};
#pragma unroll
  for (int kb = 0; kb < D; kb += 4) {
    v2f a, b0, b1;
    a[0] = X[(rowBase + mn) * D + kb + koff];
    a[1] = X[(rowBase + mn) * D + kb + koff + 1];
    b0[0] = W[(kb + koff) * D + mn];
    b0[1] = W[(kb + koff + 1) * D + mn];
    b1[0] = W[(kb + koff) * D + 16 + mn];
    b1[1] = W[(kb + koff + 1) * D + 16 + mn];
    acc0 = __builtin_amdgcn_wmma_f32_16x16x4_f32(false, a, false, b0, (short)0, acc0,
                                                 false, false);
    acc1 = __builtin_amdgcn_wmma_f32_16x16x4_f32(false, a, false, b1, (short)0, acc1,
                                                 false, false);
  }
#pragma unroll
  for (int v = 0; v < 8; ++v) {
    int row = rowBase + v + (hi ? 8 : 0);
    Y[row * D + mn] = acc0[v];
    Y[row * D + 16 + mn] = acc1[v];
  }
}

// ---------------------------------------------------------------------------
// a[n] = dot(x0_new[n,:], weff[0:32]); b[e] = dot(x1_new[e,:], weff[32:64])
// ---------------------------------------------------------------------------
__global__ void __launch_bounds__(256) k_proj_ab(const float* __restrict__ x0,
                                                 const float* __restrict__ x1,
                                                 const float* __restrict__ weff,
                                                 float* __restrict__ a,
                                                 float* __restrict__ b) {
  int i = blockIdx.x * blockDim.x + threadIdx.x;
  float sa = 0.f, sb = 0.f;
#pragma unroll
  for (int d = 0; d < D; ++d) {
    sa += x0[i * D + d] * weff[d];
    sb += x1[i * D + d] * weff[D + d];
  }
  a[i] = sa;
  b[i] = sb;
}

// ---------------------------------------------------------------------------
// out[n, e] = a[n] + b[e] + beff at every incidence nonzero.
// Duplicate pairs write identical values => matches .set semantics.
// ---------------------------------------------------------------------------
__global__ void __launch_bounds__(256) k_scatter_out(const int* __restrict__ node_idx,
                                                     const int* __restrict__ edge_idx,
                                                     const float* __restrict__ a,
                                                     const float* __restrict__ b,
                                                     const float* __restrict__ beff,
                                                     float* __restrict__ out) {
  int i = blockIdx.x * blockDim.x + threadIdx.x;
  int n = node_idx[i];
  int e = edge_idx[i];
  out[(size_t)n * N_EDGES + e] = a[n] + b[e] + beff[0];
}

extern "C" void kernel_launch(void* const* d_in, const int* in_sizes, int n_in,
                              void* d_out, int out_size, void* d_ws, size_t ws_size,
                              hipStream_t stream) {
  const float* x_0  = (const float*)d_in[0];
  // d_in[1] = incidence_1 (dense) -- intentionally unused: node_idx/edge_idx
  // carry the full sparse structure.
  const float* W_l1 = (const float*)d_in[2];
  const float* W_l2 = (const float*)d_in[3];
  const float* Wm1  = (const float*)d_in[4];
  const float* bm1  = (const float*)d_in[5];
  const float* Wm2  = (const float*)d_in[6];
  const float* bm2  = (const float*)d_in[7];
  const float* Wm3  = (const float*)d_in[8];
  const float* bm3  = (const float*)d_in[9];
  const int* node_idx = (const int*)d_in[10];
  const int* edge_idx = (const int*)d_in[11];
  float* out = (float*)d_out;

  // Workspace layout (floats): ~5.1 MB total
  float* ws   = (float*)d_ws;
  float* x1a  = ws;                        // [E*D] layer-1 edge aggregate
  float* t    = x1a  + N_EDGES * D;        // [E*D] edge linear output (reused)
  float* x0l1 = t    + N_EDGES * D;        // [N*D] layer-1 node features
  float* x1b  = x0l1 + N_NODES * D;        // [E*D] layer-2 edge agg == x_1_new
  float* x0l2 = x1b  + N_EDGES * D;        // [N*D] layer-2 node  == x_0_new
  float* avec = x0l2 + N_NODES * D;        // [N]
  float* bvec = avec + N_NODES;            // [E]
  float* weff = bvec + N_EDGES;            // [64]
  float* beff = weff + 2 * D;              // [1]

  // Bulk-zero the 256 MB dense output first (roofline floor: ~11us @ 23.3TB/s)
  hipMemsetAsync(out, 0, (size_t)out_size * sizeof(float), stream);

  // Effective affine MLP weights
  k_build_weff<<<1, 64, 0, stream>>>(Wm1, bm1, Wm2, bm2, Wm3, bm3, weff, beff);

  // ---- UniGCN layer 1 ----
  k_edge_agg<<<N_EDGES / 8, 256, 0, stream>>>(node_idx, x_0, x1a);
  k_edge_linear_wmma<<<N_EDGES / 16 / 8, 256, 0, stream>>>(x1a, W_l1, t);
  hipMemsetAsync(x0l1, 0, (size_t)N_NODES * D * sizeof(float), stream);
  k_node_scatter<<<N_EDGES / 8, 256, 0, stream>>>(node_idx, t, x0l1);

  // ---- UniGCN layer 2 ----
  k_edge_agg<<<N_EDGES / 8, 256, 0, stream>>>(node_idx, x0l1, x1b);
  k_edge_linear_wmma<<<N_EDGES / 16 / 8, 256, 0, stream>>>(x1b, W_l2, t);
  hipMemsetAsync(x0l2, 0, (size_t)N_NODES * D * sizeof(float), stream);
  k_node_scatter<<<N_EDGES / 8, 256, 0, stream>>>(node_idx, t, x0l2);

  // ---- collapsed MLP: per-node / per-edge projections, then sparse scatter
  k_proj_ab<<<N_NODES / 256, 256, 0, stream>>>(x0l2, x1b, weff, avec, bvec);
  k_scatter_out<<<NNZ_TOT / 256, 256, 0, stream>>>(node_idx, edge_idx, avec, bvec,
                                                   beff, out);
}